// MultiHeadAttentionLayer_10144712753901
// MI455X (gfx1250) — compile-verified
//
#include <hip/hip_runtime.h>

#define N_NODES 50000
#define N_EDGES 800000
#define IN_DIM 128
#define OUT_TOT 128   // OUT_DIM * NUM_HEADS
#define NUM_HEADS 8
#define LDS_STRIDE 132  // 16x128 tile padded: 4-bank row skew -> conflict-free fragment reads

static_assert(N_NODES % 16 == 0, "row tiling assumes multiple of 16");

typedef __attribute__((ext_vector_type(2))) float v2f;
typedef __attribute__((ext_vector_type(8))) float v8f;

// ---------------------------------------------------------------- zero
__global__ void gat_zero_kernel(float* __restrict__ out, float* __restrict__ z) {
    int i = blockIdx.x * blockDim.x + threadIdx.x;
    if (i < N_NODES * OUT_TOT) out[i] = 0.0f;
    if (i < N_NODES * NUM_HEADS) z[i] = 0.0f;
}

// One 16x16 output tile of O = A(16x128) * W(128x128) + b, via 32 x WMMA f32 16x16x4.
// Inlined with kernel-arg pointers directly -> global_load (no generic/flat path).
__device__ __forceinline__ void qkv_tile(const float* tA,
                                         const float* __restrict__ W,
                                         const float* __restrict__ b,
                                         float* __restrict__ O,
                                         int n_base, int lane, int row_base) {
    const int half = lane >> 4;       // 0: lanes 0-15, 1: lanes 16-31
    const int n16  = lane & 15;       // N (B/C/D) and M (A) index
    const float bias = b[n_base + n16];

    // A fragment base: lane m holds K = k0 + 2*half + {0,1}
    const float* Ab = tA + n16 * LDS_STRIDE + 2 * half;
    // B fragment base: lane n holds W[k0 + 2*half + {0,1}][n_base + n16]
    const float* Wb = W + half * (2 * OUT_TOT) + n_base + n16;

    v8f acc;
    #pragma unroll
    for (int r = 0; r < 8; ++r) acc[r] = bias;

    #pragma unroll
    for (int k0 = 0; k0 < IN_DIM; k0 += 4) {
        v2f a, bb;
        a.x  = Ab[k0];
        a.y  = Ab[k0 + 1];
        bb.x = Wb[k0 * OUT_TOT];            // imm offset k0*512 B
        bb.y = Wb[k0 * OUT_TOT + OUT_TOT];  // imm offset k0*512+512 B
        acc = __builtin_amdgcn_wmma_f32_16x16x4_f32(
            false, a, false, bb, (short)0, acc, false, false);
    }

    // C/D layout: VGPR r -> M = r + 8*half, N = n16
    float* Ob = O + (size_t)(row_base + 8 * half) * OUT_TOT + n_base + n16;
    #pragma unroll
    for (int r = 0; r < 8; ++r) Ob[r * OUT_TOT] = acc[r];
}

// ---------------------------------------------------------------- QKV projection
__global__ __launch_bounds__(256)
void gat_qkv_kernel(const float* __restrict__ h,
                    const float* __restrict__ Wq, const float* __restrict__ bq,
                    const float* __restrict__ Wk, const float* __restrict__ bk,
                    const float* __restrict__ Wv, const float* __restrict__ bv,
                    float* __restrict__ Qb, float* __restrict__ Kb, float* __restrict__ Vb) {
    __shared__ float tileA[16 * LDS_STRIDE];

    const int row_base = blockIdx.x * 16;
    const int tid = threadIdx.x;

    // Stage 16x128 A tile: thread t -> row t/16, 8 consecutive cols (no tail: 50000 % 16 == 0)
    {
        const int r = tid >> 4;
        const int c = (tid & 15) * 8;
        const float4* sp = (const float4*)(h + (size_t)(row_base + r) * IN_DIM + c);
        float4 v0 = sp[0], v1 = sp[1];
        float* dp = &tileA[r * LDS_STRIDE + c];
        dp[0] = v0.x; dp[1] = v0.y; dp[2] = v0.z; dp[3] = v0.w;
        dp[4] = v1.x; dp[5] = v1.y; dp[6] = v1.z; dp[7] = v1.w;
    }
    __syncthreads();

    const int wave   = tid >> 5;
    const int lane   = tid & 31;
    const int n_base = wave * 16;   // 8 waves <-> 8 column tiles; each wave does Q,K,V

    qkv_tile(tileA, Wq, bq, Qb, n_base, lane, row_base);
    qkv_tile(tileA, Wk, bk, Kb, n_base, lane, row_base);
    qkv_tile(tileA, Wv, bv, Vb, n_base, lane, row_base);
}

// ---------------------------------------------------------------- edge phase: one wave32 per edge
__device__ __forceinline__ void atomic_add_f32(float* p, float v) {
    __hip_atomic_fetch_add(p, v, __ATOMIC_RELAXED, __HIP_MEMORY_SCOPE_AGENT);
}

__global__ __launch_bounds__(256)
void gat_edge_kernel(const int* __restrict__ src, const int* __restrict__ dst,
                     const float* __restrict__ Qb, const float* __restrict__ Kb,
                     const float* __restrict__ Vb,
                     float* __restrict__ out, float* __restrict__ z) {
    const int tid  = threadIdx.x;
    const int lane = tid & 31;
    const int edge = blockIdx.x * 8 + (tid >> 5);
    if (edge >= N_EDGES) return;

    const int s = src[edge];
    const int d = dst[edge];
    const int c = lane * 4;                  // 4 channels per lane, 128 total

    const float4 kv = *(const float4*)(Kb + (size_t)s * OUT_TOT + c);
    const float4 qv = *(const float4*)(Qb + (size_t)d * OUT_TOT + c);
    float partial = kv.x * qv.x + kv.y * qv.y + kv.z * qv.z + kv.w * qv.w;

    // reduce the 16-wide per-head dot within each group of 4 lanes
    partial += __shfl_xor(partial, 1, 32);
    partial += __shfl_xor(partial, 2, 32);

    // score = exp(clip(dot / sqrt(16), -5, 5)); shared by the head's 4 lanes
    const float sc = __expf(fminf(fmaxf(partial * 0.25f, -5.0f), 5.0f));

    const float4 vv = *(const float4*)(Vb + (size_t)s * OUT_TOT + c);
    float* op = out + (size_t)d * OUT_TOT + c;
    atomic_add_f32(op + 0, sc * vv.x);
    atomic_add_f32(op + 1, sc * vv.y);
    atomic_add_f32(op + 2, sc * vv.z);
    atomic_add_f32(op + 3, sc * vv.w);

    if ((lane & 3) == 0) atomic_add_f32(z + (size_t)d * NUM_HEADS + (lane >> 2), sc);
}

// ---------------------------------------------------------------- normalize
__global__ void gat_norm_kernel(float* __restrict__ out, const float* __restrict__ z) {
    int i = blockIdx.x * blockDim.x + threadIdx.x;
    if (i >= N_NODES * OUT_TOT) return;
    const int n  = i >> 7;
    const int hh = (i >> 4) & 7;
    out[i] = out[i] / z[n * NUM_HEADS + hh];
}

// ---------------------------------------------------------------- launch
extern "C" void kernel_launch(void* const* d_in, const int* in_sizes, int n_in,
                              void* d_out, int out_size, void* d_ws, size_t ws_size,
                              hipStream_t stream) {
    const float* h   = (const float*)d_in[0];
    const int*   src = (const int*)d_in[1];
    const int*   dst = (const int*)d_in[2];
    const float* Wq  = (const float*)d_in[3];
    const float* bq  = (const float*)d_in[4];
    const float* Wk  = (const float*)d_in[5];
    const float* bk  = (const float*)d_in[6];
    const float* Wv  = (const float*)d_in[7];
    const float* bv  = (const float*)d_in[8];
    float* out = (float*)d_out;

    const size_t matBytes = (size_t)N_NODES * OUT_TOT * sizeof(float); // 25.6 MB
    char* ws = (char*)d_ws;
    float* Qb = (float*)(ws);
    float* Kb = (float*)(ws + matBytes);
    float* Vb = (float*)(ws + 2 * matBytes);
    float* z  = (float*)(ws + 3 * matBytes);   // 50000 x 8 floats

    gat_zero_kernel<<<(N_NODES * OUT_TOT + 255) / 256, 256, 0, stream>>>(out, z);
    gat_qkv_kernel<<<N_NODES / 16, 256, 0, stream>>>(h, Wq, bq, Wk, bk, Wv, bv,
                                                     Qb, Kb, Vb);
    gat_edge_kernel<<<N_EDGES / 8, 256, 0, stream>>>(src, dst, Qb, Kb, Vb, out, z);
    gat_norm_kernel<<<(N_NODES * OUT_TOT + 255) / 256, 256, 0, stream>>>(out, z);
}